// RefineByNocsHead_34282428957288
// MI455X (gfx1250) — compile-verified
//
#include <hip/hip_runtime.h>

#define HH 352
#define WW 1216
#define NB 16
#define KK 61
#define NH 88
#define NW 304

typedef float v2f __attribute__((ext_vector_type(2)));
typedef float v8f __attribute__((ext_vector_type(8)));

// Kernel A: bilinear-upsample nocs (3,88,304) -> (3,352,1216), subtract depth,
// pack (b0,b1,b2,mask) as float4 per pixel into workspace (6.8MB, L2-resident).
__global__ __launch_bounds__(256)
void base_kernel(const float* __restrict__ nocs,
                 const float* __restrict__ dense_depth,
                 float4* __restrict__ base) {
    int idx = blockIdx.x * blockDim.x + threadIdx.x;
    if (idx >= HH * WW) return;
    int y = idx / WW, x = idx % WW;
    // jax.image.resize bilinear, half-pixel centers, clamped at edges
    float sx = (x + 0.5f) * 0.25f - 0.5f;
    float sy = (y + 0.5f) * 0.25f - 0.5f;
    sx = fminf(fmaxf(sx, 0.0f), (float)(NW - 1));
    sy = fminf(fmaxf(sy, 0.0f), (float)(NH - 1));
    int x0 = (int)sx, y0 = (int)sy;
    float fxf = sx - (float)x0, fyf = sy - (float)y0;
    int x1 = min(x0 + 1, NW - 1), y1 = min(y0 + 1, NH - 1);

    const float* dd = dense_depth + idx * 3;
    float d0 = dd[0], d1 = dd[1], d2 = dd[2];

    float b[3];
#pragma unroll
    for (int c = 0; c < 3; ++c) {
        const float* pl = nocs + c * (NH * NW);
        float n00 = pl[y0 * NW + x0], n01 = pl[y0 * NW + x1];
        float n10 = pl[y1 * NW + x0], n11 = pl[y1 * NW + x1];
        float top = n00 + (n01 - n00) * fxf;
        float bot = n10 + (n11 - n10) * fxf;
        b[c] = top + (bot - top) * fyf;
    }
    float w = (d2 >= 1.0f) ? 1.0f : 0.0f;
    base[idx] = make_float4(b[0] - d0, b[1] - d1, b[2] - d2, w);
}

// Kernel B: one block (8 wave32) per box. Per wave chunk = 4 pixels x 64
// candidates. Lane holds candidate k = (lane&15) + 16g; B-matrix rows are the
// 4 pixels; A = ones => V_WMMA_F32_16X16X4_F32 performs the exact-f32 pixel
// reduction per candidate column, accumulated through C across chunks.
// Pixel address maintained incrementally (stride 32/lane): no integer division
// and no divergent control flow in the hot loop.
__global__ __launch_bounds__(256)
void box_kernel(const float4* __restrict__ base,
                const int* __restrict__ boxes,
                const float* __restrict__ pcenter,
                const float* __restrict__ bdepth,
                const float* __restrict__ dims,
                const float* __restrict__ intr,
                float* __restrict__ out) {
    const int n = blockIdx.x;
    const int tid = threadIdx.x;
    const int lane = tid & 31;
    const int wave = tid >> 5;

    __shared__ float part[8][64];
    __shared__ float loss[64];

    const int bx1 = boxes[n * 4 + 0], by1 = boxes[n * 4 + 1];
    const int bx2 = boxes[n * 4 + 2], by2 = boxes[n * 4 + 3];
    const int bw = bx2 - bx1;
    const int area = bw * (by2 - by1);

    const float fx = intr[0], fy = intr[4], cxi = intr[2], cyi = intr[5];
    const float ax = (pcenter[n * 2 + 0] - cxi) / fx;
    const float ay = (pcenter[n * 2 + 1] - cyi) / fy;
    const float bd = bdepth[n];

    // candidate depths for this lane's 4 candidate groups
    float dk[4];
#pragma unroll
    for (int g = 0; g < 4; ++g)
        dk[g] = bd - 3.0f + 0.1f * (float)((lane & 15) + 16 * g);

    v2f onesA;
    onesA.x = 1.0f;
    onesA.y = 1.0f;

    v8f acc[4];
#pragma unroll
    for (int g = 0; g < 4; ++g) {
#pragma unroll
        for (int i = 0; i < 8; ++i) acc[g][i] = 0.0f;
    }

    const int nchunks = (area + 3) >> 2;
    const int plo = 2 * (lane >> 4);  // lanes 0-15: rows 0,1; lanes 16-31: rows 2,3

    // Incremental trackers for the two pixels (j=0,1) this lane feeds into B
    // rows: pp = linear pixel index (validity), xx = column in box (wrap
    // detect), ai = flat index into base (address). Stride is 32 pixels per
    // wave-iteration; bw >= 32 here so at most one row-wrap per step (second
    // conditional covers bw >= 16). One div/mod at init only. Tail reads land
    // at most one row past the box (y2 <= HH-2): in-bounds, weight 0.
    int pp[2], xx[2], ai[2];
#pragma unroll
    for (int j = 0; j < 2; ++j) {
        int p0 = 4 * wave + plo + j;
        pp[j] = p0;
        xx[j] = p0 % bw;
        ai[j] = (by1 + p0 / bw) * WW + bx1 + xx[j];
    }

    for (int c = wave; c < nchunks; c += 8) {  // wave-uniform loop: EXEC all-ones
        float4 f[2];
        float wgt[2];
#pragma unroll
        for (int j = 0; j < 2; ++j) {
            f[j] = base[ai[j]];
            // unconditional use of f.w so the compiler issues one b128 load
            // (no divergent partial load of the .w component)
            wgt[j] = f[j].w * ((pp[j] < area) ? 1.0f : 0.0f);
        }
#pragma unroll
        for (int g = 0; g < 4; ++g) {
            float d = dk[g];
            v2f b;
            b.x = wgt[0] * (fabsf(fmaf(ax, d, f[0].x)) +
                            fabsf(fmaf(ay, d, f[0].y)) +
                            fabsf(d + f[0].z));
            b.y = wgt[1] * (fabsf(fmaf(ax, d, f[1].x)) +
                            fabsf(fmaf(ay, d, f[1].y)) +
                            fabsf(d + f[1].z));
            acc[g] = __builtin_amdgcn_wmma_f32_16x16x4_f32(
                false, onesA, false, b, (short)0, acc[g], false, false);
        }
        // advance trackers by 32 pixels, branchless row wrap
#pragma unroll
        for (int j = 0; j < 2; ++j) {
            pp[j] += 32;
            xx[j] += 32;
            ai[j] += 32;
            if (xx[j] >= bw) { xx[j] -= bw; ai[j] += WW - bw; }
            if (xx[j] >= bw) { xx[j] -= bw; ai[j] += WW - bw; }
        }
    }

    // D row m=0 lives in C/D VGPR0 on lanes 0-15: loss partial for candidate (16g+lane)
    if (lane < 16) {
#pragma unroll
        for (int g = 0; g < 4; ++g) part[wave][g * 16 + lane] = acc[g][0];
    }
    __syncthreads();

    if (tid < 64) {
        float s = 0.0f;
#pragma unroll
        for (int wv = 0; wv < 8; ++wv) s += part[wv][tid];
        loss[tid] = s;
    }
    __syncthreads();

    if (tid == 0) {
        int best = 0;
        float bl = loss[0];
        for (int k = 1; k < KK; ++k) {
            if (loss[k] < bl) { bl = loss[k]; best = k; }  // first-min tie-break
        }
        float d = bd - 3.0f + 0.1f * (float)best;
        out[n * 3 + 0] = ax * d;
        out[n * 3 + 1] = ay * d + 0.5f * dims[n * 3 + 1];
        out[n * 3 + 2] = d;
    }
}

extern "C" void kernel_launch(void* const* d_in, const int* in_sizes, int n_in,
                              void* d_out, int out_size, void* d_ws, size_t ws_size,
                              hipStream_t stream) {
    const float* nocs  = (const float*)d_in[0];
    const float* dd    = (const float*)d_in[1];
    const int*   boxes = (const int*)d_in[2];
    const float* pc    = (const float*)d_in[3];
    const float* bd    = (const float*)d_in[4];
    const float* dims  = (const float*)d_in[5];
    const float* intr  = (const float*)d_in[6];
    float4* base = (float4*)d_ws;   // needs HH*WW*16 = 6,848,512 bytes
    float* out = (float*)d_out;

    int npix = HH * WW;
    base_kernel<<<(npix + 255) / 256, 256, 0, stream>>>(nocs, dd, base);
    box_kernel<<<NB, 256, 0, stream>>>(base, boxes, pc, bd, dims, intr, out);
}